// Graph_OurConvNet_47648367182375
// MI455X (gfx1250) — compile-verified
//
#include <hip/hip_runtime.h>
#include <math.h>

// Problem constants (from reference)
#define N_NODES 2708
#define N_EDGES 13264
#define F_IN    1433
#define H_DIM   128
#define C_OUT   7
#define BN_EPS  1e-5f
#define NT      2          // 16-col tiles per wave in the WMMA GEMM

typedef float v2f __attribute__((ext_vector_type(2)));
typedef float v8f __attribute__((ext_vector_type(8)));

// -------------------------------------------------------------------------
// Decode dense one-hot E_start/E_end into src[e], dst[e] + in-degree per node.
// Single streaming pass over 2 * E*N floats (~288 MB -> ~12us at 23.3 TB/s).
// -------------------------------------------------------------------------
__global__ void extract_edges_kernel(const float* __restrict__ Estart,
                                     const float* __restrict__ Eend,
                                     int* __restrict__ src, int* __restrict__ dst,
                                     float* __restrict__ deg) {
  const long long total = (long long)N_EDGES * N_NODES;
  for (long long i = (long long)blockIdx.x * blockDim.x + threadIdx.x; i < total;
       i += (long long)gridDim.x * blockDim.x) {
    int e = (int)(i / N_NODES);
    int n = (int)(i % N_NODES);
    if (Estart[i] > 0.5f) src[e] = n;
    if (Eend[i]   > 0.5f) { dst[e] = n; atomicAdd(&deg[n], 1.0f); }
  }
}

// -------------------------------------------------------------------------
// out[M x 128] = X[M x K] @ W[128 x K]^T using V_WMMA_F32_16X16X4_F32.
// One wave computes a 16 x (16*NT) strip: the A fragment is loaded once per
// k-step and reused for NT independent WMMA accumulator chains.
//
// fp32 A fragment 16x4: lanes 0-15 hold rows M=0..15 with {K0,K1}; lanes
// 16-31 hold the same rows with {K2,K3}.  B fragment 4x16 mirrors this.
// C/D: VGPR j -> row j (lanes 0-15) / row j+8 (lanes 16-31), col = lane&15.
//
// No per-load predication in the main loop: invalid rows are clamped (their
// results are discarded at the guarded store), and the K remainder is one
// masked tail iteration using clamped indices + value selects.
// -------------------------------------------------------------------------
__global__ void gemm_xwt_wmma(const float* __restrict__ X,
                              const float* __restrict__ W,
                              float* __restrict__ out,
                              int M, int K) {
  const int lane = threadIdx.x & 31;
  const int lo   = lane & 15;
  const int hi   = lane >> 4;               // 0 or 1
  const int m0   = blockIdx.x * 16;
  const int h0   = blockIdx.y * (16 * NT);

  int row = m0 + lo;
  if (row >= M) row = M - 1;                // clamp: garbage rows never stored

  const float* __restrict__ xr = X + (size_t)row * K;
  const float* __restrict__ wr = W + (size_t)(h0 + lo) * K;  // tile t adds t*16*K
  const size_t wstep = (size_t)16 * K;
  const int koff = 2 * hi;

  v8f acc[NT];
  #pragma unroll
  for (int t = 0; t < NT; ++t) acc[t] = (v8f){0.f,0.f,0.f,0.f,0.f,0.f,0.f,0.f};

  int k0 = 0;
  #pragma unroll 2
  for (; k0 + 4 <= K; k0 += 4) {
    const int kk = k0 + koff;
    v2f a;
    a.x = xr[kk];
    a.y = xr[kk + 1];
    #pragma unroll
    for (int t = 0; t < NT; ++t) {
      v2f b;
      b.x = wr[(size_t)t * wstep + kk];
      b.y = wr[(size_t)t * wstep + kk + 1];
      acc[t] = __builtin_amdgcn_wmma_f32_16x16x4_f32(
          false, a, false, b, (short)0, acc[t], false, false);
    }
  }

  if (k0 < K) {                             // masked tail (1..3 valid k)
    const int kk  = k0 + koff;
    const int kc0 = (kk     < K) ? kk     : K - 1;
    const int kc1 = (kk + 1 < K) ? kk + 1 : K - 1;
    v2f a;
    a.x = (kk     < K) ? xr[kc0] : 0.0f;
    a.y = (kk + 1 < K) ? xr[kc1] : 0.0f;
    #pragma unroll
    for (int t = 0; t < NT; ++t) {
      v2f b;
      b.x = (kk     < K) ? wr[(size_t)t * wstep + kc0] : 0.0f;
      b.y = (kk + 1 < K) ? wr[(size_t)t * wstep + kc1] : 0.0f;
      acc[t] = __builtin_amdgcn_wmma_f32_16x16x4_f32(
          false, a, false, b, (short)0, acc[t], false, false);
    }
  }

  #pragma unroll
  for (int t = 0; t < NT; ++t) {
    #pragma unroll
    for (int j = 0; j < 8; ++j) {
      const int r = m0 + j + 8 * hi;
      if (r < M) out[(size_t)r * H_DIM + (h0 + 16 * t + lo)] = acc[t][j];
    }
  }
}

// -------------------------------------------------------------------------
// Edge gating: acc[dst] += sigmoid(Vix[dst] + Vjx[src] + bv) * Uix[src]
// -------------------------------------------------------------------------
__global__ void edge_gate_kernel(const float* __restrict__ Uix,
                                 const float* __restrict__ Vix,
                                 const float* __restrict__ Vjx,
                                 const float* __restrict__ bv,
                                 const int* __restrict__ src,
                                 const int* __restrict__ dst,
                                 float* __restrict__ acc) {
  const int total = N_EDGES * H_DIM;
  for (int idx = blockIdx.x * blockDim.x + threadIdx.x; idx < total;
       idx += gridDim.x * blockDim.x) {
    const int e = idx / H_DIM;
    const int h = idx - e * H_DIM;
    const int s = src[e];
    const int d = dst[e];
    const float g  = Vix[(size_t)d * H_DIM + h] + Vjx[(size_t)s * H_DIM + h] + bv[h];
    const float x1 = 1.0f / (1.0f + expf(-g));
    const float x2 = Uix[(size_t)s * H_DIM + h];
    atomicAdd(&acc[(size_t)d * H_DIM + h], x1 * x2);
  }
}

// -------------------------------------------------------------------------
// Per-node normalization: out = (acc + bu) / max(indeg, 1)
// -------------------------------------------------------------------------
__global__ void node_norm_kernel(const float* __restrict__ acc,
                                 const float* __restrict__ bu,
                                 const float* __restrict__ deg,
                                 float* __restrict__ out) {
  const int total = N_NODES * H_DIM;
  for (int idx = blockIdx.x * blockDim.x + threadIdx.x; idx < total;
       idx += gridDim.x * blockDim.x) {
    const int n = idx / H_DIM;
    const int h = idx - n * H_DIM;
    out[idx] = (acc[idx] + bu[h]) / fmaxf(deg[n], 1.0f);
  }
}

// -------------------------------------------------------------------------
// Fused BatchNorm (population mean/var over N rows, per column) with optional
// residual add and leaky-relu epilogue.  One block per feature column.
// -------------------------------------------------------------------------
__global__ void bn_kernel(const float* __restrict__ in,
                          float* __restrict__ out,
                          const float* __restrict__ g,
                          const float* __restrict__ b,
                          const float* __restrict__ resid,
                          int do_leaky) {
  __shared__ float ssum[256];
  __shared__ float ssq[256];
  const int h = blockIdx.x;
  float s = 0.f, q = 0.f;
  for (int r = threadIdx.x; r < N_NODES; r += blockDim.x) {
    const float v = in[(size_t)r * H_DIM + h];
    s += v; q += v * v;
  }
  ssum[threadIdx.x] = s; ssq[threadIdx.x] = q;
  __syncthreads();
  for (int off = 128; off > 0; off >>= 1) {
    if (threadIdx.x < off) {
      ssum[threadIdx.x] += ssum[threadIdx.x + off];
      ssq[threadIdx.x]  += ssq[threadIdx.x + off];
    }
    __syncthreads();
  }
  const float mean = ssum[0] / (float)N_NODES;
  const float var  = ssq[0] / (float)N_NODES - mean * mean;
  const float rstd = rsqrtf(var + BN_EPS);
  const float gg = g[h], bb = b[h];
  for (int r = threadIdx.x; r < N_NODES; r += blockDim.x) {
    float v = (in[(size_t)r * H_DIM + h] - mean) * rstd * gg + bb;
    if (resid) v += resid[(size_t)r * H_DIM + h];
    if (do_leaky) v = (v >= 0.f) ? v : 0.1f * v;
    out[(size_t)r * H_DIM + h] = v;
  }
}

// -------------------------------------------------------------------------
// Final FC: out[n,c] = h[n,:] . fc_w[c,:] + fc_b[c]   (C=7, tiny)
// -------------------------------------------------------------------------
__global__ void fc_kernel(const float* __restrict__ hbuf,
                          const float* __restrict__ w,
                          const float* __restrict__ bias,
                          float* __restrict__ out) {
  const int total = N_NODES * C_OUT;
  for (int idx = blockIdx.x * blockDim.x + threadIdx.x; idx < total;
       idx += gridDim.x * blockDim.x) {
    const int n = idx / C_OUT;
    const int c = idx - n * C_OUT;
    const float* __restrict__ hr = hbuf + (size_t)n * H_DIM;
    const float* __restrict__ wr = w + (size_t)c * H_DIM;
    float s = bias[c];
    #pragma unroll 8
    for (int k = 0; k < H_DIM; ++k) s += hr[k] * wr[k];
    out[idx] = s;
  }
}

// -------------------------------------------------------------------------
// Host orchestration
// -------------------------------------------------------------------------
extern "C" void kernel_launch(void* const* d_in, const int* in_sizes, int n_in,
                              void* d_out, int out_size, void* d_ws, size_t ws_size,
                              hipStream_t stream) {
  (void)in_sizes; (void)n_in; (void)out_size; (void)ws_size;

  const float* x      = (const float*)d_in[0];
  const float* Estart = (const float*)d_in[1];
  const float* Eend   = (const float*)d_in[2];
  // d_in[3]=E_identity, d_in[4]=E_dropin : unused on eval path
  const float* fc_w   = (const float*)d_in[50];
  const float* fc_b   = (const float*)d_in[51];

  // ---- carve workspace ----
  char*  ws  = (char*)d_ws;
  size_t off = 0;
  auto alloc = [&](size_t bytes) -> void* {
    void* p = ws + off;
    off += (bytes + 255) & ~(size_t)255;
    return p;
  };
  const size_t nhBytes = (size_t)N_NODES * H_DIM * sizeof(float);
  int*   srcIdx = (int*)  alloc((size_t)N_EDGES * sizeof(int));
  int*   dstIdx = (int*)  alloc((size_t)N_EDGES * sizeof(int));
  float* deg    = (float*)alloc((size_t)N_NODES * sizeof(float));
  float* Uix    = (float*)alloc(nhBytes);
  float* Vix    = (float*)alloc(nhBytes);
  float* Vjx    = (float*)alloc(nhBytes);
  float* accb   = (float*)alloc(nhBytes);
  float* t1     = (float*)alloc(nhBytes);
  float* t2     = (float*)alloc(nhBytes);
  float* resid  = (float*)alloc(nhBytes);
  float* hA     = (float*)alloc(nhBytes);
  float* hB     = (float*)alloc(nhBytes);

  // ---- decode edge one-hots (deterministic each call) ----
  hipMemsetAsync(deg, 0, (size_t)N_NODES * sizeof(float), stream);
  extract_edges_kernel<<<4096, 256, 0, stream>>>(Estart, Eend, srcIdx, dstIdx, deg);

  const dim3 ggrid((N_NODES + 15) / 16, H_DIM / (16 * NT));
  const int  eblk = 256;
  const int  egrd = (N_EDGES * H_DIM + eblk - 1) / eblk;
  const int  ngrd = (N_NODES * H_DIM + eblk - 1) / eblk;

  const float* cur = x;
  float* houts[2] = {hA, hB};

  for (int c = 0; c < 3; ++c) {
    const int base = 5 + 15 * c;
    const int K    = (c == 0) ? F_IN : H_DIM;
    const float* Ui1 = (const float*)d_in[base + 0];
    const float* Vi1 = (const float*)d_in[base + 1];
    const float* Vj1 = (const float*)d_in[base + 2];
    const float* bu1 = (const float*)d_in[base + 3];
    const float* bv1 = (const float*)d_in[base + 4];
    const float* Ui2 = (const float*)d_in[base + 5];
    const float* Vi2 = (const float*)d_in[base + 6];
    const float* Vj2 = (const float*)d_in[base + 7];
    const float* bu2 = (const float*)d_in[base + 8];
    const float* bv2 = (const float*)d_in[base + 9];
    const float* g1  = (const float*)d_in[base + 10];
    const float* b1  = (const float*)d_in[base + 11];
    const float* g2  = (const float*)d_in[base + 12];
    const float* b2  = (const float*)d_in[base + 13];
    const float* Rm  = (const float*)d_in[base + 14];

    // ---- half 1 ----
    gemm_xwt_wmma<<<ggrid, 32, 0, stream>>>(cur, Ui1, Uix,   N_NODES, K);
    gemm_xwt_wmma<<<ggrid, 32, 0, stream>>>(cur, Vi1, Vix,   N_NODES, K);
    gemm_xwt_wmma<<<ggrid, 32, 0, stream>>>(cur, Vj1, Vjx,   N_NODES, K);
    gemm_xwt_wmma<<<ggrid, 32, 0, stream>>>(cur, Rm,  resid, N_NODES, K);

    hipMemsetAsync(accb, 0, nhBytes, stream);
    edge_gate_kernel<<<egrd, eblk, 0, stream>>>(Uix, Vix, Vjx, bv1, srcIdx, dstIdx, accb);
    node_norm_kernel<<<ngrd, eblk, 0, stream>>>(accb, bu1, deg, t1);
    bn_kernel<<<H_DIM, 256, 0, stream>>>(t1, t2, g1, b1, /*resid=*/nullptr, /*leaky=*/1);

    // ---- half 2 ----
    gemm_xwt_wmma<<<ggrid, 32, 0, stream>>>(t2, Ui2, Uix, N_NODES, H_DIM);
    gemm_xwt_wmma<<<ggrid, 32, 0, stream>>>(t2, Vi2, Vix, N_NODES, H_DIM);
    gemm_xwt_wmma<<<ggrid, 32, 0, stream>>>(t2, Vj2, Vjx, N_NODES, H_DIM);

    hipMemsetAsync(accb, 0, nhBytes, stream);
    edge_gate_kernel<<<egrd, eblk, 0, stream>>>(Uix, Vix, Vjx, bv2, srcIdx, dstIdx, accb);
    node_norm_kernel<<<ngrd, eblk, 0, stream>>>(accb, bu2, deg, t1);
    // BN (no leaky inside), then + residual, then leaky — fused epilogue
    bn_kernel<<<H_DIM, 256, 0, stream>>>(t1, houts[c & 1], g2, b2, resid, /*leaky=*/1);

    cur = houts[c & 1];
  }

  fc_kernel<<<(N_NODES * C_OUT + 255) / 256, 256, 0, stream>>>(cur, fc_w, fc_b, (float*)d_out);
}